// SHGCNLayer_73641509257331
// MI455X (gfx1250) — compile-verified
//
#include <hip/hip_runtime.h>
#include <hip/hip_bf16.h>
#include <math.h>

#define EPSV 1e-5f
#define D 256
#define ROWS 32                         // rows per block (100000 % 32 == 0 -> 3125 blocks)

typedef float v2f __attribute__((ext_vector_type(2)));
typedef float v8f __attribute__((ext_vector_type(8)));

// ---------------------------------------------------------------------------
// Kernel 1: scalars + hyperbolic bias.  ws layout (floats):
//   ws[0]=c  ws[1]=sqrt_c  ws[2]=||b_hyp||^2  ws[3]=pad  ws[4..259]=b_hyp
//   ws[512...] = h_tan [N, 256]
// ---------------------------------------------------------------------------
__global__ void precompute_kernel(const float* __restrict__ bias,
                                  const float* __restrict__ c_theta,
                                  float* __restrict__ ws) {
    __shared__ float red[256];
    int t = threadIdx.x;
    float bv = bias[t];
    red[t] = bv * bv;
    __syncthreads();
    for (int s = 128; s > 0; s >>= 1) {
        if (t < s) red[t] += red[t + s];
        __syncthreads();
    }
    float sum = red[0];
    float ct = c_theta[0];
    float c  = (ct > 20.f) ? ct : log1pf(expf(ct));   // softplus
    float sc = sqrtf(c);
    float n  = fmaxf(sqrtf(sum), EPSV);
    float f  = tanhf(sc * n) / (sc * n);              // exp_map_0 factor
    ws[4 + t] = f * bv;                               // b_hyp
    if (t == 0) {
        ws[0] = c;
        ws[1] = sc;
        ws[2] = f * f * sum;                          // ||b_hyp||^2
    }
}

// ---------------------------------------------------------------------------
// Kernel 2/5: zero + relu (float4 granularity)
// ---------------------------------------------------------------------------
__global__ void zero_kernel(float4* __restrict__ out, int nvec) {
    int i = blockIdx.x * blockDim.x + threadIdx.x;
    if (i < nvec) out[i] = make_float4(0.f, 0.f, 0.f, 0.f);
}

__global__ void relu_kernel(float4* __restrict__ out, int nvec) {
    int i = blockIdx.x * blockDim.x + threadIdx.x;
    if (i < nvec) {
        float4 v = out[i];
        v.x = fmaxf(v.x, 0.f); v.y = fmaxf(v.y, 0.f);
        v.z = fmaxf(v.z, 0.f); v.w = fmaxf(v.w, 0.f);
        out[i] = v;
    }
}

// ---------------------------------------------------------------------------
// Kernel 3: fused GEMM (fp32 WMMA 16x16x4) + hyperbolic epilogue.
// Block = 512 threads = 16 waves; block computes ROWS(32) rows x all 256 cols.
// Wave w owns N-tile [16w,16w+16) and TWO M-tiles (rows 0-15, 16-31 of the
// block), so each B fragment feeds two WMMAs.  K=256 -> 128 WMMAs per wave.
// ---------------------------------------------------------------------------
__global__ __launch_bounds__(512) void fused_gemm_hyp_kernel(
        const float* __restrict__ x, const float* __restrict__ kern,
        const float* __restrict__ ws, float* __restrict__ h_tan) {
    __shared__ float xs[ROWS * D];      // 32KB: A tile, then reused as z tile
    __shared__ float rowA[ROWS], rowB[ROWS];

    const int tid  = threadIdx.x;
    const int wave = tid >> 5;          // 0..15 -> N tile
    const int lane = tid & 31;
    const int hi   = lane >> 4;         // half-wave
    const int l15  = lane & 15;
    const int row0 = blockIdx.x * ROWS;
    const int col0 = wave * 16;

    // cooperative load of 32x256 x-tile (coalesced float4)
    {
        const float4* xv  = (const float4*)(x + (size_t)row0 * D);
        float4*       xsv = (float4*)xs;
        for (int i = tid; i < ROWS * D / 4; i += 512) xsv[i] = xv[i];
    }
    __syncthreads();

    const float c_  = ws[0];
    const float sc  = ws[1];
    const float bb  = ws[2];

    // ---- GEMM: 2 M-tiles per wave, shared B fragment ---------------------
    v8f acc0 = {}, acc1 = {};
#pragma unroll 4
    for (int k0 = 0; k0 < D; k0 += 4) {
        const int ka = k0 + 2 * hi;     // K index for this half-wave
        v2f a0, a1, b;
        b.x  = kern[(size_t)ka * D + col0 + l15];        // B[ka][n]
        b.y  = kern[(size_t)(ka + 1) * D + col0 + l15];  // B[ka+1][n]
        a0.x = xs[l15 * D + ka];                         // A rows 0-15
        a0.y = xs[l15 * D + ka + 1];
        a1.x = xs[(16 + l15) * D + ka];                  // A rows 16-31
        a1.y = xs[(16 + l15) * D + ka + 1];
        acc0 = __builtin_amdgcn_wmma_f32_16x16x4_f32(
                false, a0, false, b, (short)0, acc0, false, false);
        acc1 = __builtin_amdgcn_wmma_f32_16x16x4_f32(
                false, a1, false, b, (short)0, acc1, false, false);
    }

    // ---- stage z tiles in LDS for row reductions --------------------------
    __syncthreads();                    // everyone done reading xs as A-tile
#pragma unroll
    for (int v = 0; v < 8; ++v) {
        int r = v + 8 * hi;             // C layout: VGPR v -> M = v + 8*half
        xs[r * D + col0 + l15]        = acc0[v];
        xs[(16 + r) * D + col0 + l15] = acc1[v];
    }
    __syncthreads();

    // ---- wave w reduces rows {w, w+16}: ||z||^2 and <z, b_hyp> ------------
#pragma unroll
    for (int t = 0; t < 2; ++t) {
        const int r = wave + 16 * t;
        float s0 = 0.f, s1 = 0.f;
        for (int ci = lane; ci < D; ci += 32) {
            float z  = xs[r * D + ci];
            float bh = ws[4 + ci];
            s0 += z * z;
            s1 += z * bh;
        }
        for (int off = 16; off > 0; off >>= 1) {
            s0 += __shfl_down(s0, off, 32);
            s1 += __shfl_down(s1, off, 32);
        }
        if (lane == 0) {
            // scalar hyperbolic chain for row r
            float nz2 = s0, zb = s1;
            float nz  = fmaxf(sqrtf(nz2), EPSV);
            float fz  = tanhf(sc * nz) / (sc * nz);       // exp_map factor
            float aa  = fz * fz * nz2;                    // ||a||^2
            float ab  = fz * zb;                          // <a, b_hyp>
            float al  = 1.f + 2.f * c_ * ab + c_ * bb;    // coeff on a
            float be  = 1.f - c_ * aa;                    // coeff on b
            float den = fmaxf(1.f + 2.f * c_ * ab + c_ * c_ * aa * bb, EPSV);
            float hh  = (al * al * aa + 2.f * al * be * ab + be * be * bb)
                        / (den * den);                    // ||h||^2
            float nh  = fmaxf(sqrtf(hh), EPSV);
            float mx  = (1.f - EPSV) / sc;
            float ps  = (nh > mx) ? (mx / nh) : 1.f;      // project scale
            float np  = fmaxf(ps * nh, EPSV);
            float arg = fminf(sc * np, 1.f - EPSV);
            float lf  = atanhf(arg) / (sc * np);          // log_map factor
            rowA[r] = lf * ps * (al / den) * fz;          // net coeff on z
            rowB[r] = lf * ps * (be / den);               // net coeff on b_hyp
        }
    }
    __syncthreads();

    // ---- fused store of h_tan --------------------------------------------
    const float bh = ws[4 + col0 + l15];
#pragma unroll
    for (int v = 0; v < 8; ++v) {
        int r0 = v + 8 * hi;
        int r1 = 16 + r0;
        h_tan[(size_t)(row0 + r0) * D + col0 + l15] = rowA[r0] * acc0[v] + rowB[r0] * bh;
        h_tan[(size_t)(row0 + r1) * D + col0 + l15] = rowA[r1] * acc1[v] + rowB[r1] * bh;
    }
}

// ---------------------------------------------------------------------------
// Kernel 4: edge gather + scatter-add.  One thread per (edge, float4 chunk):
// 64 chunks cover 256 cols, so `e` is wave-uniform on wave32; readfirstlane
// makes the edge-descriptor loads scalar (SMEM) instead of 32x redundant VMEM.
// ---------------------------------------------------------------------------
__global__ void edge_scatter_kernel(const float* __restrict__ h_tan,
                                    const int* __restrict__ adj_rows,
                                    const int* __restrict__ adj_cols,
                                    const float* __restrict__ adj_vals,
                                    float* __restrict__ out,
                                    long long total) {
    long long gid = (long long)blockIdx.x * blockDim.x + threadIdx.x;
    if (gid >= total) return;
    int e     = __builtin_amdgcn_readfirstlane((int)(gid >> 6)); // wave-uniform
    int chunk = (int)(gid & 63);
    int src   = adj_cols[e];
    int dst   = adj_rows[e];
    float v   = adj_vals[e];
    const float4 h = *(const float4*)(h_tan + (size_t)src * D + chunk * 4);
    float* o = out + (size_t)dst * D + chunk * 4;
    atomicAdd(o + 0, v * h.x);
    atomicAdd(o + 1, v * h.y);
    atomicAdd(o + 2, v * h.z);
    atomicAdd(o + 3, v * h.w);
}

// ---------------------------------------------------------------------------
extern "C" void kernel_launch(void* const* d_in, const int* in_sizes, int n_in,
                              void* d_out, int out_size, void* d_ws, size_t ws_size,
                              hipStream_t stream) {
    const float* x        = (const float*)d_in[0];
    const int*   adj_rows = (const int*)d_in[1];
    const int*   adj_cols = (const int*)d_in[2];
    const float* adj_vals = (const float*)d_in[3];
    const float* kern     = (const float*)d_in[4];
    const float* bias     = (const float*)d_in[5];
    const float* c_theta  = (const float*)d_in[6];
    float* out = (float*)d_out;
    float* ws  = (float*)d_ws;
    float* h_tan = ws + 512;

    const int N = in_sizes[0] / D;          // 100000 (divisible by ROWS=32)
    const long long E = in_sizes[1];        // 3.2M

    // 1: scalars + b_hyp
    precompute_kernel<<<1, 256, 0, stream>>>(bias, c_theta, ws);

    // 2: zero the (poisoned) output accumulator
    int nvec = out_size / 4;
    zero_kernel<<<(nvec + 255) / 256, 256, 0, stream>>>((float4*)out, nvec);

    // 3: fused fp32-WMMA GEMM + hyperbolic epilogue -> h_tan in scratch
    fused_gemm_hyp_kernel<<<N / ROWS, 512, 0, stream>>>(x, kern, ws, h_tan);

    // 4: edge gather/scatter
    long long total = E * 64;
    long long nblk  = (total + 255) / 256;
    edge_scatter_kernel<<<dim3((unsigned)nblk), 256, 0, stream>>>(
        h_tan, adj_rows, adj_cols, adj_vals, out, total);

    // 5: relu in place
    relu_kernel<<<(nvec + 255) / 256, 256, 0, stream>>>((float4*)out, nvec);
}